// SabiaNetwork_89867895701753
// MI455X (gfx1250) — compile-verified
//
#include <hip/hip_runtime.h>
#include <math.h>

// Problem constants (match reference)
#define NN 50000
#define EE 1600000
#define CC 32
#define BB 10
#define GG 16
#define HH 100
#define NC (NN * CC)          // 1,600,000 floats per (N,32) plane
#define RMAXF 3.5f
#define INVK 0.17677669529663687f   // 1/sqrt(32)
#define SQRT3 1.7320508075688772f
#define SQRTB 3.1622776601683795f   // sqrt(10)

#define EW 4                  // waves per block in edge kernels
#define HSTRIDE 116           // LDS staging stride for h tile (conflict-free)

typedef float v2f __attribute__((ext_vector_type(2)));
typedef float v8f __attribute__((ext_vector_type(8)));

// ---------------------------------------------------------------------------
// Kernel 0: zero accumulators (aggS, aggV[3], agg2 = first 5*NC floats of ws)
// and zero d_out (512 floats) since we accumulate into it atomically.
// ---------------------------------------------------------------------------
__global__ __launch_bounds__(256) void k_zero(float* ws, float* out) {
    int gid = blockIdx.x * 256 + threadIdx.x;
    if (gid < 5 * NC) ws[gid] = 0.0f;
    if (gid < GG * CC) out[gid] = 0.0f;
}

// ---------------------------------------------------------------------------
// Device helpers (fast rcp -> single v_rcp_f32 instead of IEEE div expansion)
// ---------------------------------------------------------------------------
__device__ __forceinline__ float silu_f(float z) {
    return z * __builtin_amdgcn_rcpf(1.0f + __expf(-z));
}
__device__ __forceinline__ float sigmoid_f(float z) {
    return __builtin_amdgcn_rcpf(1.0f + __expf(-z));
}
// radial basis value for (possibly padded) index b; zero for b >= BB
__device__ __forceinline__ float emb_at(float r, int b) {
    const float step = RMAXF / (float)(BB + 1);
    const float istep = (float)(BB + 1) / RMAXF;
    float diff = (r - (float)(b + 1) * step) * istep;
    bool in = (diff > -1.0f) && (diff < 1.0f) && (b < BB);
    return in ? __cosf(0.5f * (float)M_PI * diff) * SQRTB : 0.0f;
}

// ---------------------------------------------------------------------------
// WMMA radial MLP: per wave, 16 edges.
//   layer1: emb(16x10 pad 12) @ wa(pad 12x112) -> h(16x112), silu, LDS stage
//   layer2: h(16x100) @ wb(100x64)             -> wD[4] (16x64 in D layout)
// ---------------------------------------------------------------------------
__device__ __forceinline__ void radial_mlp_wmma(float r, int m, int kh,
                                                const float* lds_wa,
                                                const float* lds_wb,
                                                float* lds_hw,  // this wave's 16*HSTRIDE
                                                v8f wD[4]) {
    int k0 = 2 * kh;
    v2f a0, a1, a2;
    a0.x = emb_at(r, k0);     a0.y = emb_at(r, k0 + 1);
    a1.x = emb_at(r, 4 + k0); a1.y = emb_at(r, 5 + k0);
    a2.x = emb_at(r, 8 + k0); a2.y = emb_at(r, 9 + k0);

    // layer 1: 7 column tiles of 16 (100 valid cols, rest pad to zero)
#pragma unroll
    for (int ct = 0; ct < 7; ++ct) {
        int n = ct * 16 + m;
        v8f acc = {0.f, 0.f, 0.f, 0.f, 0.f, 0.f, 0.f, 0.f};
        v2f b;
        b.x = lds_wa[(k0 + 0) * 112 + n];
        b.y = lds_wa[(k0 + 1) * 112 + n];
        acc = __builtin_amdgcn_wmma_f32_16x16x4_f32(false, a0, false, b,
                                                    (short)0, acc, false, false);
        b.x = lds_wa[(k0 + 4) * 112 + n];
        b.y = lds_wa[(k0 + 5) * 112 + n];
        acc = __builtin_amdgcn_wmma_f32_16x16x4_f32(false, a1, false, b,
                                                    (short)0, acc, false, false);
        b.x = lds_wa[(k0 + 8) * 112 + n];
        b.y = lds_wa[(k0 + 9) * 112 + n];
        acc = __builtin_amdgcn_wmma_f32_16x16x4_f32(false, a2, false, b,
                                                    (short)0, acc, false, false);
        // silu + stage to LDS in A-friendly (row-major) layout
#pragma unroll
        for (int j = 0; j < 8; ++j)
            lds_hw[(j + 8 * kh) * HSTRIDE + n] = silu_f(acc[j]);
    }

    // layer 2: K = 100 (25 k-steps), 4 column tiles of 16
#pragma unroll
    for (int ct = 0; ct < 4; ++ct) {
        int n = ct * 16 + m;
        v8f acc = {0.f, 0.f, 0.f, 0.f, 0.f, 0.f, 0.f, 0.f};
#pragma unroll
        for (int kk = 0; kk < 25; ++kk) {
            int k = 4 * kk + k0;
            v2f a = *(const v2f*)&lds_hw[m * HSTRIDE + k];
            v2f b;
            b.x = lds_wb[k * 64 + n];
            b.y = lds_wb[(k + 1) * 64 + n];
            acc = __builtin_amdgcn_wmma_f32_16x16x4_f32(false, a, false, b,
                                                        (short)0, acc, false, false);
        }
        wD[ct] = acc;
    }
}

// ---------------------------------------------------------------------------
// Kernel 1: edge pass 1 (WMMA MLP).  One wave = 16 edges, 4 waves/block.
// Gather x[src]; atomic scatter to aggS (N,32) and aggV (3 planes of N,32).
// ---------------------------------------------------------------------------
__global__ __launch_bounds__(128) void k_edge1(const float* __restrict__ edge_vec,
                                               const int* __restrict__ edge_src,
                                               const int* __restrict__ edge_dst,
                                               const float* __restrict__ x,
                                               const float* __restrict__ w1a,
                                               const float* __restrict__ w1b,
                                               float* __restrict__ aggS,
                                               float* __restrict__ aggV) {
    __shared__ float lds_wa[12 * 112];          // padded 12x112
    __shared__ float lds_wb[HH * 64];           // 100x64
    __shared__ float lds_h[EW][16 * HSTRIDE];   // per-wave h staging
    __shared__ float lds_y[EW][16 * 3];
    __shared__ int   lds_src[EW][16];
    __shared__ int   lds_dst[EW][16];

    for (int i = threadIdx.x; i < 12 * 112; i += 128) {
        int k = i / 112, n = i % 112;
        lds_wa[i] = (k < BB && n < HH) ? w1a[k * HH + n] : 0.0f;
    }
    for (int i = threadIdx.x; i < HH * 64; i += 128) lds_wb[i] = w1b[i];
    __syncthreads();

    int wave = threadIdx.x >> 5;
    int lane = threadIdx.x & 31;
    int m = lane & 15, kh = lane >> 4;
    int base = (blockIdx.x * EW + wave) * 16;
    if (base >= EE) return;  // uniform per wave (EE % 64 == 0 anyway)

    int e = base + m;
    float ev0 = edge_vec[3 * e + 0];
    float ev1 = edge_vec[3 * e + 1];
    float ev2 = edge_vec[3 * e + 2];
    float r = sqrtf(ev0 * ev0 + ev1 * ev1 + ev2 * ev2);
    float ir = __builtin_amdgcn_rcpf(fmaxf(r, 1e-12f));
    if (kh == 0) {
        lds_y[wave][m * 3 + 0] = SQRT3 * ev0 * ir;
        lds_y[wave][m * 3 + 1] = SQRT3 * ev1 * ir;
        lds_y[wave][m * 3 + 2] = SQRT3 * ev2 * ir;
        lds_src[wave][m] = edge_src[e];
        lds_dst[wave][m] = edge_dst[e];
    }

    v8f wD[4];
    radial_mlp_wmma(r, m, kh, lds_wa, lds_wb, &lds_h[wave][0], wD);

    // scatter: for fixed (j,kh) all 16 lanes handle the SAME edge at
    // consecutive channels -> coalesced gathers and atomics.
#pragma unroll
    for (int j = 0; j < 8; ++j) {
        int el = j + 8 * kh;
        int sb = lds_src[wave][el] * CC;
        int db = lds_dst[wave][el] * CC;
        float yy0 = lds_y[wave][el * 3 + 0];
        float yy1 = lds_y[wave][el * 3 + 1];
        float yy2 = lds_y[wave][el * 3 + 2];
        float xs0 = x[sb + m];
        float xs1 = x[sb + 16 + m];
        atomicAdd(&aggS[db + m],      xs0 * wD[0][j] * INVK);
        atomicAdd(&aggS[db + 16 + m], xs1 * wD[1][j] * INVK);
        float mv0 = xs0 * wD[2][j] * INVK;
        float mv1 = xs1 * wD[3][j] * INVK;
        atomicAdd(&aggV[0 * NC + db + m],      mv0 * yy0);
        atomicAdd(&aggV[1 * NC + db + m],      mv0 * yy1);
        atomicAdd(&aggV[2 * NC + db + m],      mv0 * yy2);
        atomicAdd(&aggV[0 * NC + db + 16 + m], mv1 * yy0);
        atomicAdd(&aggV[1 * NC + db + 16 + m], mv1 * yy1);
        atomicAdd(&aggV[2 * NC + db + 16 + m], mv1 * yy2);
    }
}

// ---------------------------------------------------------------------------
// Kernel 2: node pass 1 (WMMA f32 16x16x4).
// h_s = [aggS | x] @ [[lin_s1],[sc1]]  (K=64, Ncols=64)
// s = silu(h_s[:, :32]) -> sbuf ; g = sigmoid(h_s[:, 32:])
// h_v[d] = aggV[d] @ lin_v1 (K=32) ; v[d] = h_v[d] * g -> vbuf
// ---------------------------------------------------------------------------
__global__ __launch_bounds__(256) void k_node1(const float* __restrict__ aggS,
                                               const float* __restrict__ aggV,
                                               const float* __restrict__ x,
                                               const float* __restrict__ lin_s1,
                                               const float* __restrict__ lin_v1,
                                               const float* __restrict__ sc1,
                                               float* __restrict__ sbuf,
                                               float* __restrict__ vbuf) {
    __shared__ float lds_B[64 * 64];  // [k][n]: k<32 -> lin_s1, else sc1
    __shared__ float lds_V[32 * 32];  // lin_v1[c][e]
    for (int i = threadIdx.x; i < 64 * 64; i += 256) {
        int k = i >> 6, n = i & 63;
        lds_B[i] = (k < 32) ? lin_s1[k * 64 + n] : sc1[(k - 32) * 64 + n];
    }
    for (int i = threadIdx.x; i < 32 * 32; i += 256) lds_V[i] = lin_v1[i];
    __syncthreads();

    int wave = threadIdx.x >> 5;
    int lane = threadIdx.x & 31;
    int tile = blockIdx.x * 8 + wave;
    if (tile * 16 >= NN) return;

    int m = lane & 15;
    int kh = lane >> 4;
    int row = tile * 16 + m;

    v2f a[16];
#pragma unroll
    for (int kk = 0; kk < 8; ++kk) {
        int k = 4 * kk + 2 * kh;
        a[kk] = *(const v2f*)(aggS + row * CC + k);
    }
#pragma unroll
    for (int kk = 0; kk < 8; ++kk) {
        int k = 4 * kk + 2 * kh;
        a[8 + kk] = *(const v2f*)(x + row * CC + k);
    }

    v8f hs[4];
#pragma unroll
    for (int ct = 0; ct < 4; ++ct) {
        v8f acc = {0.f, 0.f, 0.f, 0.f, 0.f, 0.f, 0.f, 0.f};
#pragma unroll
        for (int kk = 0; kk < 16; ++kk) {
            int k = 4 * kk + 2 * kh;
            v2f b;
            b.x = lds_B[k * 64 + ct * 16 + m];
            b.y = lds_B[(k + 1) * 64 + ct * 16 + m];
            acc = __builtin_amdgcn_wmma_f32_16x16x4_f32(
                false, a[kk], false, b, (short)0, acc, false, false);
        }
        hs[ct] = acc;
    }

#pragma unroll
    for (int ct = 0; ct < 2; ++ct) {
#pragma unroll
        for (int j = 0; j < 8; ++j) {
            int rr = tile * 16 + j + 8 * kh;
            sbuf[rr * CC + ct * 16 + m] = silu_f(hs[ct][j]);
        }
    }
    v8f g[2];
#pragma unroll
    for (int ct = 0; ct < 2; ++ct)
#pragma unroll
        for (int j = 0; j < 8; ++j) g[ct][j] = sigmoid_f(hs[2 + ct][j]);

#pragma unroll
    for (int d = 0; d < 3; ++d) {
        v2f av[8];
#pragma unroll
        for (int kk = 0; kk < 8; ++kk) {
            int k = 4 * kk + 2 * kh;
            av[kk] = *(const v2f*)(aggV + d * NC + row * CC + k);
        }
#pragma unroll
        for (int ct = 0; ct < 2; ++ct) {
            v8f acc = {0.f, 0.f, 0.f, 0.f, 0.f, 0.f, 0.f, 0.f};
#pragma unroll
            for (int kk = 0; kk < 8; ++kk) {
                int k = 4 * kk + 2 * kh;
                v2f b;
                b.x = lds_V[k * 32 + ct * 16 + m];
                b.y = lds_V[(k + 1) * 32 + ct * 16 + m];
                acc = __builtin_amdgcn_wmma_f32_16x16x4_f32(
                    false, av[kk], false, b, (short)0, acc, false, false);
            }
#pragma unroll
            for (int j = 0; j < 8; ++j) {
                int rr = tile * 16 + j + 8 * kh;
                vbuf[d * NC + rr * CC + ct * 16 + m] = acc[j] * g[ct][j];
            }
        }
    }
}

// ---------------------------------------------------------------------------
// Kernel 3: edge pass 2 (WMMA MLP).  Gather s[src], v[src];
// m = s*w0 + (v . Y1)*wd ; atomic scatter into agg2.
// ---------------------------------------------------------------------------
__global__ __launch_bounds__(128) void k_edge2(const float* __restrict__ edge_vec,
                                               const int* __restrict__ edge_src,
                                               const int* __restrict__ edge_dst,
                                               const float* __restrict__ sbuf,
                                               const float* __restrict__ vbuf,
                                               const float* __restrict__ w2a,
                                               const float* __restrict__ w2b,
                                               float* __restrict__ agg2) {
    __shared__ float lds_wa[12 * 112];
    __shared__ float lds_wb[HH * 64];
    __shared__ float lds_h[EW][16 * HSTRIDE];
    __shared__ float lds_y[EW][16 * 3];
    __shared__ int   lds_src[EW][16];
    __shared__ int   lds_dst[EW][16];

    for (int i = threadIdx.x; i < 12 * 112; i += 128) {
        int k = i / 112, n = i % 112;
        lds_wa[i] = (k < BB && n < HH) ? w2a[k * HH + n] : 0.0f;
    }
    for (int i = threadIdx.x; i < HH * 64; i += 128) lds_wb[i] = w2b[i];
    __syncthreads();

    int wave = threadIdx.x >> 5;
    int lane = threadIdx.x & 31;
    int m = lane & 15, kh = lane >> 4;
    int base = (blockIdx.x * EW + wave) * 16;
    if (base >= EE) return;

    int e = base + m;
    float ev0 = edge_vec[3 * e + 0];
    float ev1 = edge_vec[3 * e + 1];
    float ev2 = edge_vec[3 * e + 2];
    float r = sqrtf(ev0 * ev0 + ev1 * ev1 + ev2 * ev2);
    float ir = __builtin_amdgcn_rcpf(fmaxf(r, 1e-12f));
    if (kh == 0) {
        lds_y[wave][m * 3 + 0] = SQRT3 * ev0 * ir;
        lds_y[wave][m * 3 + 1] = SQRT3 * ev1 * ir;
        lds_y[wave][m * 3 + 2] = SQRT3 * ev2 * ir;
        lds_src[wave][m] = edge_src[e];
        lds_dst[wave][m] = edge_dst[e];
    }

    v8f wD[4];
    radial_mlp_wmma(r, m, kh, lds_wa, lds_wb, &lds_h[wave][0], wD);

#pragma unroll
    for (int j = 0; j < 8; ++j) {
        int el = j + 8 * kh;
        int sb = lds_src[wave][el] * CC;
        int db = lds_dst[wave][el] * CC;
        float yy0 = lds_y[wave][el * 3 + 0];
        float yy1 = lds_y[wave][el * 3 + 1];
        float yy2 = lds_y[wave][el * 3 + 2];
        float sv0 = sbuf[sb + m];
        float sv1 = sbuf[sb + 16 + m];
        float dv0 = vbuf[0 * NC + sb + m] * yy0 +
                    vbuf[1 * NC + sb + m] * yy1 +
                    vbuf[2 * NC + sb + m] * yy2;
        float dv1 = vbuf[0 * NC + sb + 16 + m] * yy0 +
                    vbuf[1 * NC + sb + 16 + m] * yy1 +
                    vbuf[2 * NC + sb + 16 + m] * yy2;
        atomicAdd(&agg2[db + m],      (sv0 * wD[0][j] + dv0 * wD[2][j]) * INVK);
        atomicAdd(&agg2[db + 16 + m], (sv1 * wD[1][j] + dv1 * wD[3][j]) * INVK);
    }
}

// ---------------------------------------------------------------------------
// Kernel 4: node pass 2 (WMMA) + graph segment-sum via LDS (ds_add_f32).
// node_out = [agg2 | s] @ [[lin_s2],[sc2]]  (K=64, Ncols=32)
// ---------------------------------------------------------------------------
__global__ __launch_bounds__(256) void k_node2(const float* __restrict__ agg2,
                                               const float* __restrict__ sbuf,
                                               const float* __restrict__ lin_s2,
                                               const float* __restrict__ sc2,
                                               const int* __restrict__ batch,
                                               float* __restrict__ out) {
    __shared__ float lds_B[64 * 32];
    __shared__ float lds_out[GG * CC];
    for (int i = threadIdx.x; i < 64 * 32; i += 256) {
        int k = i >> 5, n = i & 31;
        lds_B[i] = (k < 32) ? lin_s2[k * 32 + n] : sc2[(k - 32) * 32 + n];
    }
    for (int i = threadIdx.x; i < GG * CC; i += 256) lds_out[i] = 0.0f;
    __syncthreads();

    int wave = threadIdx.x >> 5;
    int lane = threadIdx.x & 31;
    int tile = blockIdx.x * 8 + wave;
    if (tile * 16 < NN) {
        int m = lane & 15;
        int kh = lane >> 4;
        int row = tile * 16 + m;

        v2f a[16];
#pragma unroll
        for (int kk = 0; kk < 8; ++kk) {
            int k = 4 * kk + 2 * kh;
            a[kk] = *(const v2f*)(agg2 + row * CC + k);
        }
#pragma unroll
        for (int kk = 0; kk < 8; ++kk) {
            int k = 4 * kk + 2 * kh;
            a[8 + kk] = *(const v2f*)(sbuf + row * CC + k);
        }

        int bix[8];
#pragma unroll
        for (int j = 0; j < 8; ++j) bix[j] = batch[tile * 16 + j + 8 * kh];

#pragma unroll
        for (int ct = 0; ct < 2; ++ct) {
            v8f acc = {0.f, 0.f, 0.f, 0.f, 0.f, 0.f, 0.f, 0.f};
#pragma unroll
            for (int kk = 0; kk < 16; ++kk) {
                int k = 4 * kk + 2 * kh;
                v2f b;
                b.x = lds_B[k * 32 + ct * 16 + m];
                b.y = lds_B[(k + 1) * 32 + ct * 16 + m];
                acc = __builtin_amdgcn_wmma_f32_16x16x4_f32(
                    false, a[kk], false, b, (short)0, acc, false, false);
            }
#pragma unroll
            for (int j = 0; j < 8; ++j)
                atomicAdd(&lds_out[bix[j] * CC + ct * 16 + m], acc[j]);
        }
    }
    __syncthreads();
    for (int i = threadIdx.x; i < GG * CC; i += 256)
        atomicAdd(&out[i], lds_out[i]);
}

// ---------------------------------------------------------------------------
// Launch
// ---------------------------------------------------------------------------
extern "C" void kernel_launch(void* const* d_in, const int* in_sizes, int n_in,
                              void* d_out, int out_size, void* d_ws, size_t ws_size,
                              hipStream_t stream) {
    const float* x        = (const float*)d_in[0];
    const float* edge_vec = (const float*)d_in[1];
    const int*   edge_src = (const int*)d_in[2];
    const int*   edge_dst = (const int*)d_in[3];
    const int*   batch    = (const int*)d_in[4];
    const float* w1a      = (const float*)d_in[5];
    const float* w1b      = (const float*)d_in[6];
    const float* w2a      = (const float*)d_in[7];
    const float* w2b      = (const float*)d_in[8];
    const float* lin_s1   = (const float*)d_in[9];
    const float* lin_v1   = (const float*)d_in[10];
    const float* sc1      = (const float*)d_in[11];
    const float* lin_s2   = (const float*)d_in[12];
    const float* sc2      = (const float*)d_in[13];
    float* out = (float*)d_out;
    float* ws  = (float*)d_ws;

    float* aggS = ws;                 // N*32
    float* aggV = ws + 1 * NC;        // 3 * N*32 (d-major planes)
    float* agg2 = ws + 4 * NC;        // N*32
    float* sbuf = ws + 5 * NC;        // N*32
    float* vbuf = ws + 6 * NC;        // 3 * N*32 (d-major planes)

    // K0: zero accumulators + output
    k_zero<<<(5 * NC + 255) / 256, 256, 0, stream>>>(ws, out);

    // K1: edge pass 1 (WMMA MLP), 64 edges / block
    k_edge1<<<(EE + 63) / 64, 128, 0, stream>>>(edge_vec, edge_src, edge_dst,
                                                x, w1a, w1b, aggS, aggV);

    // K2: node pass 1 (WMMA)
    int tiles = NN / 16;  // 3125
    k_node1<<<(tiles + 7) / 8, 256, 0, stream>>>(aggS, aggV, x, lin_s1, lin_v1,
                                                 sc1, sbuf, vbuf);

    // K3: edge pass 2 (WMMA MLP)
    k_edge2<<<(EE + 63) / 64, 128, 0, stream>>>(edge_vec, edge_src, edge_dst,
                                                sbuf, vbuf, w2a, w2b, agg2);

    // K4: node pass 2 (WMMA) + graph reduction
    k_node2<<<(tiles + 7) / 8, 256, 0, stream>>>(agg2, sbuf, lin_s2, sc2,
                                                 batch, out);
}